// GPT2Attention_15556371546456
// MI455X (gfx1250) — compile-verified
//
#include <hip/hip_runtime.h>
#include <math.h>

#define HIDDEN   2048
#define NHEADS   16
#define HDIM     128
#define BSIZE    64
#define NSEQS    64
#define MAXBLK   32
#define PARTS    8
#define POS_PER_PART 256          // 4 KV blocks of 64 positions per partition
#define LDW_ATTN (3 * HIDDEN)     // W_attn leading dim (row-major [2048, 6144])

__constant__ const float kScale = 0.08838834764831845f;  // 128^-0.5

typedef __attribute__((ext_vector_type(2))) float v2f;
typedef __attribute__((ext_vector_type(8))) float v8f;

// ---------------------------------------------------------------------------
// GEMM: C[64 x 2048] = A[64 x 2048] * B[2048 x LDB] (cols 0..2047) + bias
// LDB is a compile-time constant so B row strides fold into immediate offsets.
// grid.x = 64 N-tiles of 32 (two 16-wide WMMA tiles per wave, shared A frag);
// block = 128 threads = 4 waves; wave w owns M-tile w (M=64 total).
// Uses V_WMMA_F32_16X16X4_F32 (full f32 precision, matches reference).
// ---------------------------------------------------------------------------
template <int LDB>
__global__ __launch_bounds__(128) void gemm64x2048_wmma(
    const float* __restrict__ A, const float* __restrict__ B,
    const float* __restrict__ bias, float* __restrict__ C)
{
  const int lane  = threadIdx.x & 31;
  const int wave  = threadIdx.x >> 5;
  const int mBase = wave * 16;
  const int nBase = blockIdx.x * 32;      // two adjacent 16-wide tiles
  const int nCol  = lane & 15;            // B/C: N striped across lane halves
  const int kOff  = (lane >> 4) * 2;      // lanes 16-31 hold K=+2,+3

  v8f c0 = {}, c1 = {};
  const float* __restrict__ arow = A + (mBase + (lane & 15)) * HIDDEN + kOff;
  const float* __restrict__ bcol = B + (long)kOff * LDB + nBase + nCol;

#pragma unroll 4
  for (int kb = 0; kb < HIDDEN; kb += 4) {
    v2f a, b0, b1;
    a.x = arow[kb];
    a.y = arow[kb + 1];
    const float* bp = bcol + (long)kb * LDB;
    b0.x = bp[0];
    b0.y = bp[LDB];
    b1.x = bp[16];
    b1.y = bp[LDB + 16];
    // (neg_a, A, neg_b, B, c_mod, C, reuse_a, reuse_b)
    c0 = __builtin_amdgcn_wmma_f32_16x16x4_f32(false, a, false, b0, (short)0,
                                               c0, false, false);
    c1 = __builtin_amdgcn_wmma_f32_16x16x4_f32(false, a, false, b1, (short)0,
                                               c1, false, false);
  }

  const int rowHalf = (lane >> 4) * 8;    // C: VGPR i = rows i / i+8 by lane half
  const float bv0 = bias[nBase + nCol];
  const float bv1 = bias[nBase + 16 + nCol];
#pragma unroll
  for (int i = 0; i < 8; ++i) {
    const int m = mBase + rowHalf + i;
    C[m * HIDDEN + nBase + nCol]      = c0[i] + bv0;
    C[m * HIDDEN + nBase + 16 + nCol] = c1[i] + bv1;
  }
}

// ---------------------------------------------------------------------------
// Flash-decode partial pass. grid = (NSEQS, PARTS); block = 256 (8 waves).
// Each wave handles 2 heads over this partition's <=256 positions with online
// softmax. K/V rows are 512B contiguous -> one float4 per lane, coalesced.
// Partials (m, l, acc[128]) go to workspace.
// ---------------------------------------------------------------------------
__global__ __launch_bounds__(256) void attn_partial(
    const float* __restrict__ q, const float* __restrict__ kc,
    const float* __restrict__ vc, const int* __restrict__ btab,
    const int* __restrict__ slen, float* __restrict__ pm,
    float* __restrict__ pl, float* __restrict__ pacc)
{
  const int seq  = blockIdx.x;
  const int part = blockIdx.y;
  const int lane = threadIdx.x & 31;
  const int wave = threadIdx.x >> 5;
  const int L      = slen[seq];
  const int pStart = part * POS_PER_PART;
  const int pEnd   = min(pStart + POS_PER_PART, L);

  for (int hsub = 0; hsub < 2; ++hsub) {
    const int head = wave * 2 + hsub;
    const float4 q4 =
        *(const float4*)(q + seq * HIDDEN + head * HDIM + lane * 4);

    float  m = -INFINITY, l = 0.f;
    float4 acc = make_float4(0.f, 0.f, 0.f, 0.f);

    for (int pos = pStart; pos < pEnd; ++pos) {
      const int  blk  = btab[seq * MAXBLK + (pos >> 6)];
      const long base = (((long)blk * BSIZE + (pos & 63)) * NHEADS + head) *
                            HDIM + lane * 4;
      const float4 k4 = *(const float4*)(kc + base);
      const float4 v4 = *(const float4*)(vc + base);
      // prefetch next position's row (same block; gfx1250 global_prefetch_b8)
      __builtin_prefetch(kc + base + NHEADS * HDIM, 0, 0);
      __builtin_prefetch(vc + base + NHEADS * HDIM, 0, 0);

      float s = q4.x * k4.x + q4.y * k4.y + q4.z * k4.z + q4.w * k4.w;
      s += __shfl_xor(s, 1);
      s += __shfl_xor(s, 2);
      s += __shfl_xor(s, 4);
      s += __shfl_xor(s, 8);
      s += __shfl_xor(s, 16);
      s *= kScale;

      const float mN    = fmaxf(m, s);
      const float alpha = __expf(m - mN);   // first iter: exp(-inf)=0
      const float p     = __expf(s - mN);
      l = l * alpha + p;
      acc.x = fmaf(p, v4.x, acc.x * alpha);
      acc.y = fmaf(p, v4.y, acc.y * alpha);
      acc.z = fmaf(p, v4.z, acc.z * alpha);
      acc.w = fmaf(p, v4.w, acc.w * alpha);
      m = mN;
    }

    const long pidx = (long)(seq * NHEADS + head) * PARTS + part;
    if (lane == 0) { pm[pidx] = m; pl[pidx] = l; }
    *(float4*)(pacc + pidx * HDIM + lane * 4) = acc;
  }
}

// ---------------------------------------------------------------------------
// Combine partitions. grid = NSEQS*NHEADS; block = 128 (one thread per dim).
// ---------------------------------------------------------------------------
__global__ __launch_bounds__(128) void attn_reduce(
    const float* __restrict__ pm, const float* __restrict__ pl,
    const float* __restrict__ pacc, float* __restrict__ attn)
{
  const int sh = blockIdx.x;    // seq*16 + head
  const int d  = threadIdx.x;   // 0..127

  float M = -INFINITY;
#pragma unroll
  for (int p = 0; p < PARTS; ++p)
    if (pl[sh * PARTS + p] > 0.f) M = fmaxf(M, pm[sh * PARTS + p]);

  float lt = 0.f, num = 0.f;
#pragma unroll
  for (int p = 0; p < PARTS; ++p) {
    const float lp = pl[sh * PARTS + p];
    if (lp > 0.f) {
      const float w = __expf(pm[sh * PARTS + p] - M);
      lt  += lp * w;
      num += w * pacc[(long)(sh * PARTS + p) * HDIM + d];
    }
  }
  const int seq = sh >> 4, head = sh & 15;
  attn[seq * HIDDEN + head * HDIM + d] = (lt > 0.f) ? num / lt : 0.f;
}

// ---------------------------------------------------------------------------
extern "C" void kernel_launch(void* const* d_in, const int* in_sizes, int n_in,
                              void* d_out, int out_size, void* d_ws,
                              size_t ws_size, hipStream_t stream)
{
  (void)in_sizes; (void)n_in; (void)out_size; (void)ws_size;
  const float* hidden  = (const float*)d_in[0];
  const float* kv      = (const float*)d_in[1];   // [2, 2048, 64, 16, 128]
  const float* W_attn  = (const float*)d_in[2];   // [2048, 6144]
  const float* b_attn  = (const float*)d_in[3];
  const float* W_proj  = (const float*)d_in[4];   // [2048, 2048]
  const float* b_proj  = (const float*)d_in[5];
  const int*   btab    = (const int*)d_in[6];     // [64, 32]
  const int*   slen    = (const int*)d_in[7];     // [64]
  float*       out     = (float*)d_out;

  const long   kvHalf  = 2048L * BSIZE * NHEADS * HDIM;  // 268435456 elems
  const float* kc = kv;
  const float* vc = kv + kvHalf;

  // workspace layout (floats): q[64*2048] | attn[64*2048] | pm[8192] | pl[8192]
  //                            | pacc[64*16*8*128]  (~5.1 MB total)
  float* q    = (float*)d_ws;
  float* attn = q + NSEQS * HIDDEN;
  float* pm   = attn + NSEQS * HIDDEN;
  float* pl   = pm + NSEQS * NHEADS * PARTS;
  float* pacc = pl + NSEQS * NHEADS * PARTS;

  // 1) Q projection (only q slice of qkv is ever used by the reference)
  gemm64x2048_wmma<LDW_ATTN><<<dim3(HIDDEN / 32), dim3(128), 0, stream>>>(
      hidden, W_attn, b_attn, q);

  // 2) paged flash-decode partials
  attn_partial<<<dim3(NSEQS, PARTS), dim3(256), 0, stream>>>(
      q, kc, vc, btab, slen, pm, pl, pacc);

  // 3) combine partitions
  attn_reduce<<<dim3(NSEQS * NHEADS), dim3(128), 0, stream>>>(pm, pl, pacc,
                                                              attn);

  // 4) output projection
  gemm64x2048_wmma<HIDDEN><<<dim3(HIDDEN / 32), dim3(128), 0, stream>>>(
      attn, W_proj, b_proj, out);
}